// UnbiasedIWKnowledgeDistillationLoss_20512763806281
// MI455X (gfx1250) — compile-verified
//
#include <hip/hip_runtime.h>
#include <math.h>

// Problem constants (from reference): inputs [8,20,512,512] f32, targets [8,16,512,512] f32
#define B_    8
#define C_    20
#define OLD_  16
#define HW_   (512 * 512)
#define NPIX_ ((size_t)B_ * HW_)            // 2,097,152
#define NT_   ((size_t)B_ * OLD_ * HW_)     // 33,554,432

typedef float v2f __attribute__((ext_vector_type(2)));
typedef float v8f __attribute__((ext_vector_type(8)));

// ws layout:
//   [0, 8)      double loss accumulator
//   [16, 656)   float hist[8][20]
//   [704, 1216) float w[8][16]

__global__ void k_zero(double* __restrict__ acc, float* __restrict__ hist) {
    if (threadIdx.x == 0) *acc = 0.0;
    if (threadIdx.x < B_ * C_) hist[threadIdx.x] = 0.0f;
}

// Pass 1: per-batch histogram of targets (torch.histc semantics: bins=20 over [0,19],
// width = 19/20 = 0.95, out-of-range ignored, clamp to last bin).
__global__ void k_hist(const float* __restrict__ t, float* __restrict__ hist) {
    __shared__ unsigned int sh[B_ * C_];
    for (int i = threadIdx.x; i < B_ * C_; i += blockDim.x) sh[i] = 0u;
    __syncthreads();
    const size_t stride = (size_t)gridDim.x * blockDim.x;
    for (size_t i = (size_t)blockIdx.x * blockDim.x + threadIdx.x; i < NT_; i += stride) {
        float v = t[i];
        if (v >= 0.0f && v <= 19.0f) {
            int bin = (int)floorf(v / 0.95f);
            bin = bin < 0 ? 0 : (bin > 19 ? 19 : bin);
            int b = (int)(i / ((size_t)OLD_ * HW_));
            atomicAdd(&sh[b * C_ + bin], 1u);   // ds_add_u32
        }
    }
    __syncthreads();
    for (int i = threadIdx.x; i < B_ * C_; i += blockDim.x) {
        unsigned int v = sh[i];
        if (v) atomicAdd(&hist[i], (float)v);   // counts < 2^24 -> exact in f32
    }
}

// Pass 2: class weights. hist==0 -> 1 (all bins), fold bins 16..19 into bin 0,
// zero bins 16.., w = (h / sum)^0.1 for first 16 bins.
__global__ void k_weights(const float* __restrict__ hist, float* __restrict__ w) {
    __shared__ float h[C_];
    __shared__ float h0s, sums;
    const int b = blockIdx.x, c = threadIdx.x;
    if (c < C_) {
        float v = hist[b * C_ + c];
        h[c] = (v == 0.0f) ? 1.0f : v;
    }
    __syncthreads();
    if (c == 0) {
        float h0 = h[0] + h[16] + h[17] + h[18] + h[19];
        float s = h0;
        for (int i = 1; i < OLD_; ++i) s += h[i];
        h0s = h0; sums = s;
    }
    __syncthreads();
    if (c < OLD_) {
        float hv = (c == 0) ? h0s : h[c];
        w[b * OLD_ + c] = powf(hv / sums, 0.1f);
    }
}

// Pass 3: fused loss. Each thread handles pixels p, p+stride, ... (stride == HW_ here,
// so channel loads stay fully coalesced and batch advances by 1 per iteration).
// Wave-level reduction done with V_WMMA_F32_16X16X4_F32 (exact fp32, B = all-ones).
__global__ void __launch_bounds__(256)
k_loss(const float* __restrict__ in, const float* __restrict__ tg,
       const float* __restrict__ w, double* __restrict__ acc) {
    const size_t stride = (size_t)gridDim.x * blockDim.x;
    float lacc = 0.0f;
    for (size_t p = (size_t)blockIdx.x * blockDim.x + threadIdx.x; p < NPIX_; p += stride) {
        const int    b  = (int)(p / HW_);
        const size_t hw = p % HW_;
        const float* ib = in + (size_t)b * C_ * HW_ + hw;
        const float* tb = tg + (size_t)b * OLD_ * HW_ + hw;
        const float* wb = w + b * OLD_;

        float x[C_];
        #pragma unroll
        for (int c = 0; c < C_; ++c) x[c] = ib[(size_t)c * HW_];

        // den = logsumexp over all 20 channels
        float m = x[0];
        #pragma unroll
        for (int c = 1; c < C_; ++c) m = fmaxf(m, x[c]);
        float s = 0.0f;
        #pragma unroll
        for (int c = 0; c < C_; ++c) s += expf(x[c] - m);
        const float den = m + logf(s);

        // outputs_bkg = logsumexp over {0,16,17,18,19} - den
        float mb = x[0];
        #pragma unroll
        for (int c = OLD_; c < C_; ++c) mb = fmaxf(mb, x[c]);
        float sb = expf(x[0] - mb);
        #pragma unroll
        for (int c = OLD_; c < C_; ++c) sb += expf(x[c] - mb);
        const float ob = mb + logf(sb) - den;

        // softmax over 16 target channels
        float tv[OLD_];
        #pragma unroll
        for (int c = 0; c < OLD_; ++c) tv[c] = tb[(size_t)c * HW_];
        float mt = tv[0];
        #pragma unroll
        for (int c = 1; c < OLD_; ++c) mt = fmaxf(mt, tv[c]);
        float st = 0.0f;
        #pragma unroll
        for (int c = 0; c < OLD_; ++c) st += expf(tv[c] - mt);
        const float inv = 1.0f / st;

        float lp = wb[0] * (expf(tv[0] - mt) * inv) * ob;
        #pragma unroll
        for (int c = 1; c < OLD_; ++c)
            lp += wb[c] * (expf(tv[c] - mt) * inv) * (x[c] - den);
        lacc += lp;
    }

    // Wave32 reduction via WMMA: A[m,0]=v_m, A[m,2]=v_{m+16}, A[*,1]=A[*,3]=0, B=ones.
    // D[m,n] = v_m + v_{m+16}; lane 0 sums D[0..7,0], lane 16 sums D[8..15,0].
    v2f a;  a[0] = lacc; a[1] = 0.0f;
    v2f bo; bo[0] = 1.0f; bo[1] = 1.0f;
    v8f cm = {};
    cm = __builtin_amdgcn_wmma_f32_16x16x4_f32(false, a, false, bo, (short)0, cm,
                                               false, false);
    float wsum = cm[0] + cm[1] + cm[2] + cm[3] + cm[4] + cm[5] + cm[6] + cm[7];
    const unsigned lane = threadIdx.x & 31u;
    if (lane == 0u || lane == 16u) atomicAdd(acc, (double)wsum);
}

__global__ void k_final(const double* __restrict__ acc, float* __restrict__ out) {
    if (threadIdx.x == 0 && blockIdx.x == 0)
        out[0] = (float)(-(*acc) / (double)NPIX_);
}

extern "C" void kernel_launch(void* const* d_in, const int* in_sizes, int n_in,
                              void* d_out, int out_size, void* d_ws, size_t ws_size,
                              hipStream_t stream) {
    const float* inp = (const float*)d_in[0];   // [8,20,512,512]
    const float* tgt = (const float*)d_in[1];   // [8,16,512,512]
    float* out = (float*)d_out;

    char*   ws   = (char*)d_ws;
    double* acc  = (double*)ws;
    float*  hist = (float*)(ws + 16);
    float*  w    = (float*)(ws + 704);

    k_zero   <<<1,    256, 0, stream>>>(acc, hist);
    k_hist   <<<2048, 256, 0, stream>>>(tgt, hist);
    k_weights<<<B_,   32,  0, stream>>>(hist, w);
    k_loss   <<<1024, 256, 0, stream>>>(inp, tgt, w, acc);
    k_final  <<<1,    32,  0, stream>>>(acc, out);
}